// MoeModel_23639499997494
// MI455X (gfx1250) — compile-verified
//
#include <hip/hip_runtime.h>

typedef float v2f __attribute__((ext_vector_type(2)));
typedef float v8f __attribute__((ext_vector_type(8)));

#define DM   16
#define NEXP 8
#define LDP  17   // padded LDS row stride (dodge bank conflicts)

__global__ __launch_bounds__(64) void moe_init_ws(float* ws) {
    int t = threadIdx.x;
    if (t < 16) ws[t] = 0.f;
}

__global__ __launch_bounds__(64) void moe_finalize(const float* __restrict__ ws,
                                                   float* __restrict__ out, int n) {
    if (threadIdx.x == 0 && blockIdx.x == 0) {
        float acc = 0.f;
        #pragma unroll
        for (int e = 0; e < NEXP; ++e) acc += ws[e] * ws[8 + e];
        float fn = (float)n;
        out[(size_t)n * 4] = 8.f * acc / (fn * fn);
    }
}

__global__ __launch_bounds__(256, 1) void moe_main(
    const float* __restrict__ x,        const float* __restrict__ w_embed,
    const float* __restrict__ b_embed,  const float* __restrict__ gate_w,
    const float* __restrict__ expert_w, const float* __restrict__ expert_b,
    const float* __restrict__ resid_w,  const float* __restrict__ resid_b,
    const float* __restrict__ comb_w,   const float* __restrict__ comb_b,
    const float* __restrict__ proj_w,   const float* __restrict__ proj_b,
    float* __restrict__ out, float* __restrict__ ws, int n)
{
    // one multi-purpose tile buffer per wave: h -> gate logits -> y (DS is
    // in-order within a wave, so sequential reuse is safe)
    __shared__ float s_t[8][DM * LDP];
    __shared__ float s_f0[8][16];               // gate_sel*cw0 per row
    __shared__ float s_f1[8][16];               // cw1 per row
    __shared__ int   s_idx[8][16];              // argmax per row
    __shared__ __align__(16) float s_o[8][DM * 4];  // out staging
    __shared__ float s_eb[NEXP * 4 * 64];       // expert B pre-gathered per-lane
    __shared__ float s_red[16];

    const int tid  = threadIdx.x;
    const int lane = tid & 31;
    const int wv   = tid >> 5;          // 8 waves per block (wave32)
    const int l    = lane & 15;
    const int h2   = (lane >> 4) << 1;  // A/B K offset: 0 or 2
    const int mh   = (lane >> 4) << 3;  // C/D row offset: 0 or 8

    if (tid < 16) s_red[tid] = 0.f;

    // pre-gather expert weights into per-lane B-layout: s_eb[(e*4+s)*64 + lane*2 + j]
    {
        const int e   = tid >> 5;       // one expert per wave's worth of threads
        const int gl  = tid & 31;
        const int gll = gl & 15;
        const int gh2 = (gl >> 4) << 1;
        #pragma unroll
        for (int s = 0; s < 4; ++s) {
            s_eb[(e * 4 + s) * 64 + gl * 2 + 0] = expert_w[e * DM * DM + (4 * s + gh2 + 0) * DM + gll];
            s_eb[(e * 4 + s) * 64 + gl * 2 + 1] = expert_w[e * DM * DM + (4 * s + gh2 + 1) * DM + gll];
        }
    }
    __syncthreads();

    // ---------------- small weights in registers (WMMA B layout) ----------------
    // B (KxN): VGPR j, lane-half hh holds B[k0 + j + 2*hh][N = lane&15]
    v2f web;                                        // embed: 4x16
    web.x = w_embed[(h2 + 0) * DM + l];
    web.y = w_embed[(h2 + 1) * DM + l];
    float bemb  = b_embed[l];
    float rbias = resid_b[l];

    v2f rb[4], gb[4], pb[4];
    float ebias[NEXP];
    #pragma unroll
    for (int s = 0; s < 4; ++s) {
        const int k0 = 4 * s + h2;
        rb[s].x = resid_w[(k0 + 0) * DM + l];
        rb[s].y = resid_w[(k0 + 1) * DM + l];
        // gate (cols 0-7) + comb (cols 8-9) fused B, zero-padded
        float g0 = 0.f, g1 = 0.f;
        if (l < NEXP)       { g0 = gate_w[(k0 + 0) * NEXP + l]; g1 = gate_w[(k0 + 1) * NEXP + l]; }
        else if (l == 8)    { g0 = comb_w[(k0 + 0) * 2 + 0];    g1 = comb_w[(k0 + 1) * 2 + 0]; }
        else if (l == 9)    { g0 = comb_w[(k0 + 0) * 2 + 1];    g1 = comb_w[(k0 + 1) * 2 + 1]; }
        gb[s].x = g0; gb[s].y = g1;
        // proj (cols 0-3), zero-padded
        pb[s].x = (l < 4) ? proj_w[(k0 + 0) * 4 + l] : 0.f;
        pb[s].y = (l < 4) ? proj_w[(k0 + 1) * 4 + l] : 0.f;
    }
    #pragma unroll
    for (int e = 0; e < NEXP; ++e) ebias[e] = expert_b[e * DM + l];

    const float gcinit = (l == 8) ? comb_b[0] : ((l == 9) ? comb_b[1] : 0.f);
    const float pbv    = proj_b[l & 3];
    const float pcinit = (l < 4) ? pbv : 0.f;

    float gsum[NEXP], csum[NEXP];   // aux accumulators (meaningful on lanes 0-15)
    #pragma unroll
    for (int e = 0; e < NEXP; ++e) { gsum[e] = 0.f; csum[e] = 0.f; }

    const int ntiles = n >> 4;
    const int nwaves = gridDim.x * 8;
    const int lelem  = (l << 2) + h2;   // per-lane element offset into a 16-token x tile

    int tile = blockIdx.x * 8 + wv;     // wave-uniform

    // prime the x-tile pipeline (wave-uniform guard)
    v2f a_cur;
    if (tile < ntiles) {
        const int t0 = __builtin_amdgcn_readfirstlane(tile) << 4;
        const float* xr = x + (size_t)t0 * 4 + lelem;
        a_cur.x = xr[0]; a_cur.y = xr[1];
    }

    for (; tile < ntiles; tile += nwaves) {
        const int tok0 = __builtin_amdgcn_readfirstlane(tile) << 4;

        // ---- embed: h = X(16x4) @ We(4x16) + b   (one WMMA, K=4)
        v8f hc;
        #pragma unroll
        for (int j = 0; j < 8; ++j) hc[j] = bemb;
        hc = __builtin_amdgcn_wmma_f32_16x16x4_f32(false, a_cur, false, web,
                                                   (short)0, hc, false, false);

        // prefetch next tile's x (hides HBM latency behind the WMMA chains)
        v2f a_next;
        {
            int nt  = tile + nwaves;
            int ntc = (nt < ntiles) ? nt : tile;     // clamp on last iteration
            const int t1 = __builtin_amdgcn_readfirstlane(ntc) << 4;
            const float* xr = x + (size_t)t1 * 4 + lelem;
            a_next.x = xr[0]; a_next.y = xr[1];
        }

        // spill h (C/D layout: row M=j+mh, col N=l) to LDS row-major
        #pragma unroll
        for (int j = 0; j < 8; ++j) s_t[wv][(j + mh) * LDP + l] = hc[j];

        // rebuild h as 4 A-layout K-slices
        v2f ha[4];
        #pragma unroll
        for (int s = 0; s < 4; ++s) {
            ha[s].x = s_t[wv][l * LDP + 4 * s + h2 + 0];
            ha[s].y = s_t[wv][l * LDP + 4 * s + h2 + 1];
        }

        // ---- fused router+comb logits: G = h @ [gate_w | comb_w | 0]
        v8f gacc;
        #pragma unroll
        for (int j = 0; j < 8; ++j) gacc[j] = gcinit;
        #pragma unroll
        for (int s = 0; s < 4; ++s)
            gacc = __builtin_amdgcn_wmma_f32_16x16x4_f32(false, ha[s], false, gb[s],
                                                         (short)0, gacc, false, false);

        // spill gate logits over the h tile (h already captured in ha regs)
        #pragma unroll
        for (int j = 0; j < 8; ++j) s_t[wv][(j + mh) * LDP + l] = gacc[j];

        // ---- residual expert: 16x16x16 (overlaps gating latency)
        v8f racc;
        #pragma unroll
        for (int j = 0; j < 8; ++j) racc[j] = rbias;
        #pragma unroll
        for (int s = 0; s < 4; ++s)
            racc = __builtin_amdgcn_wmma_f32_16x16x4_f32(false, ha[s], false, rb[s],
                                                         (short)0, racc, false, false);

        // ---- per-token softmax/argmax (lanes 0-15, token t = l)
        if (lane < 16) {
            float g[10];
            #pragma unroll
            for (int k = 0; k < 10; ++k) g[k] = s_t[wv][l * LDP + k];
            float mx = g[0]; int idx = 0;
            #pragma unroll
            for (int e = 1; e < NEXP; ++e)
                if (g[e] > mx) { mx = g[e]; idx = e; }     // first-max tie-break
            float ex[NEXP], den = 0.f;
            #pragma unroll
            for (int e = 0; e < NEXP; ++e) { ex[e] = __expf(g[e] - mx); den += ex[e]; }
            float inv = 1.f / den;
            #pragma unroll
            for (int e = 0; e < NEXP; ++e) {
                gsum[e] += ex[e] * inv;
                csum[e] += (idx == e) ? 1.f : 0.f;
            }
            float gsel = inv;                               // softmax value at argmax
            float cw0  = 1.f / (1.f + __expf(g[9] - g[8])); // softmax over 2
            s_f0[wv][l]  = gsel * cw0;
            s_f1[wv][l]  = 1.f - cw0;
            s_idx[wv][l] = idx;
        }

        // only the argmax indices are needed during the expert loop
        int idxr[8];
        #pragma unroll
        for (int j = 0; j < 8; ++j) idxr[j] = s_idx[wv][j + mh];

        // ---- dense 8-expert compute, masked select (EXEC all-1 around WMMA)
        // expert B slices come from LDS each use (volatile: keep them out of VGPRs)
        v8f moe;
        #pragma unroll
        for (int j = 0; j < 8; ++j) moe[j] = 0.f;
        #pragma unroll
        for (int e = 0; e < NEXP; ++e) {
            v8f ec;
            #pragma unroll
            for (int j = 0; j < 8; ++j) ec[j] = ebias[e];
            #pragma unroll
            for (int s = 0; s < 4; ++s) {
                v2f ebv = *(const volatile v2f*)&s_eb[(e * 4 + s) * 64 + (lane << 1)];
                ec = __builtin_amdgcn_wmma_f32_16x16x4_f32(false, ha[s], false, ebv,
                                                           (short)0, ec, false, false);
            }
            #pragma unroll
            for (int j = 0; j < 8; ++j) moe[j] = (idxr[j] == e) ? ec[j] : moe[j];
        }

        // combine factors fetched only now (keeps them out of the expert loop's
        // live range)
        float f0r[8], f1r[8];
        #pragma unroll
        for (int j = 0; j < 8; ++j) {
            f0r[j] = s_f0[wv][j + mh];
            f1r[j] = s_f1[wv][j + mh];
        }

        // ---- y = moe*gate_sel*cw0 + resid*cw1 ; spill row-major (reuse s_t)
        #pragma unroll
        for (int j = 0; j < 8; ++j)
            s_t[wv][(j + mh) * LDP + l] = moe[j] * f0r[j] + racc[j] * f1r[j];

        // ---- proj: out16x4 = y @ [proj_w | 0] + proj_b (4 WMMAs)
        v2f ya[4];
        #pragma unroll
        for (int s = 0; s < 4; ++s) {
            ya[s].x = s_t[wv][l * LDP + 4 * s + h2 + 0];
            ya[s].y = s_t[wv][l * LDP + 4 * s + h2 + 1];
        }
        v8f oc;
        #pragma unroll
        for (int j = 0; j < 8; ++j) oc[j] = pcinit;
        #pragma unroll
        for (int s = 0; s < 4; ++s)
            oc = __builtin_amdgcn_wmma_f32_16x16x4_f32(false, ya[s], false, pb[s],
                                                       (short)0, oc, false, false);

        // stage valid cols 0-3 to LDS, then one dense b128 store per 16 lanes
        if (l < 4) {
            #pragma unroll
            for (int j = 0; j < 8; ++j) s_o[wv][(j + mh) * 4 + l] = oc[j];
        }
        if (lane < 16) {
            float4 ov = *(const float4*)&s_o[wv][l * 4];
            *(float4*)(out + (size_t)(tok0 + l) * 4) = ov;
        }

        a_cur = a_next;
    }

    // ---- aux-loss reduction: regs -> LDS atomics -> global atomics
    if (lane < 16) {
        #pragma unroll
        for (int e = 0; e < NEXP; ++e) {
            atomicAdd(&s_red[e],     gsum[e]);
            atomicAdd(&s_red[8 + e], csum[e]);
        }
    }
    __syncthreads();
    if (tid < 16) atomicAdd(&ws[tid], s_red[tid]);
}

extern "C" void kernel_launch(void* const* d_in, const int* in_sizes, int n_in,
                              void* d_out, int out_size, void* d_ws, size_t ws_size,
                              hipStream_t stream) {
    const float* x        = (const float*)d_in[0];
    const float* w_embed  = (const float*)d_in[1];
    const float* b_embed  = (const float*)d_in[2];
    const float* gate_w   = (const float*)d_in[3];
    const float* expert_w = (const float*)d_in[4];
    const float* expert_b = (const float*)d_in[5];
    const float* resid_w  = (const float*)d_in[6];
    const float* resid_b  = (const float*)d_in[7];
    const float* comb_w   = (const float*)d_in[8];
    const float* comb_b   = (const float*)d_in[9];
    const float* proj_w   = (const float*)d_in[10];
    const float* proj_b   = (const float*)d_in[11];
    float* out = (float*)d_out;
    float* ws  = (float*)d_ws;

    const int n      = in_sizes[0] / 4;     // tokens
    const int ntiles = n >> 4;
    int blocks = (ntiles + 63) / 64;        // ~8 tiles per wave
    if (blocks > 1024) blocks = 1024;
    if (blocks < 1)    blocks = 1;

    moe_init_ws<<<1, 64, 0, stream>>>(ws);
    moe_main<<<blocks, 256, 0, stream>>>(x, w_embed, b_embed, gate_w, expert_w,
                                         expert_b, resid_w, resid_b, comb_w, comb_b,
                                         proj_w, proj_b, out, ws, n);
    moe_finalize<<<1, 64, 0, stream>>>(ws, out, n);
}